// E3SchNetInteraction_41901700940488
// MI455X (gfx1250) — compile-verified
//
#include <hip/hip_runtime.h>
#include <math.h>

// ---------------------------------------------------------------------------
// E3SchNet interaction block for MI455X (gfx1250), fp32, WMMA 16x16x4.
//   xs = x[:, :C] @ W_in2f_s / sqrt(C)
//   xv[n,f,i] = sum_c x[n,C+3c+i] W_in2f_v[c,f] / sqrt(C)
//   h  = ssp(f_ij @ Wf1 + bf1)
//   W  = (h @ Wf2[:, :512] + bf2[:512]) * rcut     (cols 512:768 dead downstream)
//   per-edge messages m_s + m_v -> deterministic CSR segment sum -> agg[n,1024]
//   S1 = ssp(agg[:, :256] @ W1_s / 16);  V1[n,c,i] = agg-view @ W1_v / 16
//   out[:, :128] = S1 @ W2_s / sqrt(128);  out[:,128+3c+i] = V1[:,:,i] @ W2_v / sqrt(128)
// All GEMMs run on v_wmma_f32_16x16x4_f32; each wave register-blocks a 16x64
// output strip (4 accumulator tiles) so one A fetch feeds 4 WMMAs. All N are
// multiples of 64 so the inner loop is branch-free straight-line code.
// ---------------------------------------------------------------------------

#define N_ATOMS 5000
#define N_EDGES 50000
#define CDIM 128
#define FDIM 128
#define NRBF 20

typedef __attribute__((ext_vector_type(2))) float v2f;
typedef __attribute__((ext_vector_type(8))) float v8f;

#define LOG2F_C 0.69314718055994531f
#define Y0_C 0.28209479177387814f       // 1/sqrt(4 pi)
#define Y1S_C 0.48860251190291992f      // sqrt(3/(4 pi))
#define INV_SQRT3_C 0.57735026918962576f
#define INV_SQRT128_C 0.08838834764831845f

__device__ __forceinline__ float ssp_act(float x) {
  float sp = (x > 20.0f) ? x : log1pf(expf(x));
  return sp - LOG2F_C;
}

// ---------------------------------------------------------------------------
// fp32 GEMM on V_WMMA_F32_16X16X4_F32, epilogue fused:
//   C[m,n] = act( alpha*sum_k A[m*ldAr+k*LDAC]*B[k*ldB+n] + bias[n] ) * rowscale[m]
// One wave32 -> 16x64 strip (4 WMMA tiles); 8 waves stack along M -> block
// covers 128 rows x 64 cols. grid = (ceil(M/128), N/64); requires N % 64 == 0.
// KDIM is compile-time (20/128/256) so the k-loop unrolls branch-free.
// ISA 7.12.2 f32 operand layouts:
//   A 16x4 : lane<16 -> {K=0,K=1}, lane>=16 -> {K=2,K=3}; M = lane&15
//   B 4x16 : lane<16 -> rows {K=0,K=1}, lane>=16 -> {K=2,K=3}; N = lane&15
//   C 16x16: vgpr r -> M = r + (lane<16 ? 0 : 8), N = lane&15
// LDAC==1 lets the A pair merge into one global_load_b64; LDAC==3 serves the
// stride-3 einsum views.
// ---------------------------------------------------------------------------
template <int LDAC, int KDIM>
__global__ __launch_bounds__(256) void gemm_wmma_f32(
    const float* __restrict__ A, int ldAr,
    const float* __restrict__ B, int ldB,
    const float* __restrict__ bias,      // per-column, may be null
    const float* __restrict__ rowscale,  // per-row, may be null
    float* __restrict__ Cmat, int ldCr, int ldCc,
    int M, float alpha, int act)
{
  const int lane = threadIdx.x & 31;
  const int wave = threadIdx.x >> 5;
  const int m0 = blockIdx.x * 128 + wave * 16;
  if (m0 >= M) return;               // wave-uniform
  const int n0 = blockIdx.y * 64;
  const int half = lane >> 4;        // 0: lanes 0-15, 1: lanes 16-31
  const int l16 = lane & 15;
  const int mA = m0 + l16;
  const bool mok = mA < M;
  // Clamp OOB A rows: garbage only affects C rows that are never stored.
  const int koff = half << 1;
  const float* Ap = A + (size_t)(mok ? mA : (M - 1)) * (size_t)ldAr + koff * LDAC;
  const float* Bp = B + (size_t)koff * ldB + n0 + l16;

  v8f acc0 = {}, acc1 = {}, acc2 = {}, acc3 = {};
#pragma unroll
  for (int k = 0; k < KDIM; k += 4) {
    v2f av, b0, b1, b2, b3;
    av[0] = Ap[0];
    av[1] = Ap[LDAC];                // b64-merged when LDAC==1
    b0[0] = Bp[0];   b0[1] = Bp[ldB];
    b1[0] = Bp[16];  b1[1] = Bp[ldB + 16];
    b2[0] = Bp[32];  b2[1] = Bp[ldB + 32];
    b3[0] = Bp[48];  b3[1] = Bp[ldB + 48];
    Ap += 4 * LDAC;
    Bp += (size_t)4 * ldB;
    acc0 = __builtin_amdgcn_wmma_f32_16x16x4_f32(false, av, false, b0, (short)0, acc0, false, false);
    acc1 = __builtin_amdgcn_wmma_f32_16x16x4_f32(false, av, false, b1, (short)0, acc1, false, false);
    acc2 = __builtin_amdgcn_wmma_f32_16x16x4_f32(false, av, false, b2, (short)0, acc2, false, false);
    acc3 = __builtin_amdgcn_wmma_f32_16x16x4_f32(false, av, false, b3, (short)0, acc3, false, false);
  }

  auto epilogue = [&](const v8f& acc, int t) {
    const int nB = n0 + t * 16 + l16;
    const float bb = bias ? bias[nB] : 0.0f;
    float* Ccol = Cmat + (size_t)nB * ldCc;
#pragma unroll
    for (int r = 0; r < 8; ++r) {
      const int mr = m0 + r + (half << 3);
      if (mr < M) {
        float v = acc[r] * alpha + bb;
        if (act) v = ssp_act(v);
        if (rowscale) v *= rowscale[mr];
        Ccol[(size_t)mr * ldCr] = v;
      }
    }
  };
  epilogue(acc0, 0);
  epilogue(acc1, 1);
  epilogue(acc2, 2);
  epilogue(acc3, 3);
}

// ---------------------------------------------------------------------------
// CSR row starts from sorted idx_i: start[a] = lower_bound(idx_i, a).
// start[N_ATOMS] == N_EDGES falls out automatically.
// ---------------------------------------------------------------------------
__global__ __launch_bounds__(128) void build_starts(
    const int* __restrict__ idx_i, int nE, int nA, int* __restrict__ start)
{
  const int a = blockIdx.x * blockDim.x + threadIdx.x;
  if (a > nA) return;
  int lo = 0, hi = nE;
  while (lo < hi) {
    int mid = (lo + hi) >> 1;
    if (idx_i[mid] < a) lo = mid + 1; else hi = mid;
  }
  start[a] = lo;
}

// ---------------------------------------------------------------------------
// Fused gather + spherical featurize + filter-multiply + segment sum.
// One block per atom, thread f owns channel f; 8 live accumulators:
//   agg[a, 2f]         = sum_e sj*Y0          * W[e, 2f]
//   agg[a, 2f+1]       = sum_e (vj.Yv)/sqrt3  * W[e, 2f+1]
//   agg[a, 256+6f+i]   = sum_e sj*Yv_i        * W[e, 256+2f]
//   agg[a, 256+6f+3+i] = sum_e vj_i*Y0        * W[e, 256+2f+1]
// (m_w / m_t message blocks are dead downstream and skipped.)
// Deterministic: edges visited in index order, no atomics.
// ---------------------------------------------------------------------------
__global__ __launch_bounds__(128) void edge_aggregate(
    const int* __restrict__ start, const int* __restrict__ idx_j,
    const float* __restrict__ r_ij,
    const float* __restrict__ xs,    // [n, 128]
    const float* __restrict__ xv,    // [n, 128, 3]
    const float* __restrict__ Wmat,  // [e, 512]
    float* __restrict__ agg)         // [n, 1024]
{
  const int a = blockIdx.x;
  const int f = threadIdx.x;
  const int e0 = start[a];
  const int e1 = start[a + 1];

  float as0 = 0.f, as1 = 0.f;
  float ax0 = 0.f, ax1 = 0.f, ax2 = 0.f;   // sj * Yv_i * wv0
  float ab0 = 0.f, ab1 = 0.f, ab2 = 0.f;   // vj_i * Y0 * wv1

  for (int e = e0; e < e1; ++e) {
    const int j = idx_j[e];
    const float rx = r_ij[3 * e + 0];
    const float ry = r_ij[3 * e + 1];
    const float rz = r_ij[3 * e + 2];
    const float inv = rsqrtf(rx * rx + ry * ry + rz * rz);
    const float Yx = Y1S_C * rx * inv;
    const float Yy = Y1S_C * ry * inv;
    const float Yz = Y1S_C * rz * inv;

    const float sj = xs[(size_t)j * FDIM + f];
    const float* vp = xv + (size_t)j * (3 * FDIM) + 3 * f;
    const float vx = vp[0], vy = vp[1], vz = vp[2];

    // W row is 2048B-aligned; adjacent (2f, 2f+1) pairs -> b64 loads
    const float2* Wp2 = (const float2*)(Wmat + (size_t)e * 512);
    const float2 wsv = Wp2[f];          // { W[2f], W[2f+1] }
    const float2 wvv = Wp2[128 + f];    // { W[256+2f], W[256+2f+1] }

    const float dot = vx * Yx + vy * Yy + vz * Yz;
    as0 += sj * Y0_C * wsv.x;
    as1 += dot * INV_SQRT3_C * wsv.y;
    ax0 += sj * Yx * wvv.x;
    ax1 += sj * Yy * wvv.x;
    ax2 += sj * Yz * wvv.x;
    ab0 += vx * Y0_C * wvv.y;
    ab1 += vy * Y0_C * wvv.y;
    ab2 += vz * Y0_C * wvv.y;
  }

  float* ap = agg + (size_t)a * 1024;
  ap[2 * f] = as0;
  ap[2 * f + 1] = as1;
  ap[256 + 6 * f + 0] = ax0;
  ap[256 + 6 * f + 1] = ax1;
  ap[256 + 6 * f + 2] = ax2;
  ap[256 + 6 * f + 3] = ab0;
  ap[256 + 6 * f + 4] = ab1;
  ap[256 + 6 * f + 5] = ab2;
}

// ---------------------------------------------------------------------------
extern "C" void kernel_launch(void* const* d_in, const int* in_sizes, int n_in,
                              void* d_out, int out_size, void* d_ws, size_t ws_size,
                              hipStream_t stream) {
  (void)in_sizes; (void)n_in; (void)out_size; (void)ws_size;

  const float* x        = (const float*)d_in[0];
  const int*   idx_i    = (const int*)d_in[1];
  const int*   idx_j    = (const int*)d_in[2];
  const float* f_ij     = (const float*)d_in[3];
  const float* rcut_ij  = (const float*)d_in[4];
  const float* r_ij     = (const float*)d_in[5];
  const float* W_in2f_s = (const float*)d_in[6];
  const float* W_in2f_v = (const float*)d_in[7];
  const float* Wf1      = (const float*)d_in[8];
  const float* bf1      = (const float*)d_in[9];
  const float* Wf2      = (const float*)d_in[10];
  const float* bf2      = (const float*)d_in[11];
  const float* W1_s     = (const float*)d_in[12];
  const float* W1_v     = (const float*)d_in[13];
  const float* W2_s     = (const float*)d_in[14];
  const float* W2_v     = (const float*)d_in[15];
  float* out = (float*)d_out;

  // Workspace layout (floats): ~169 MB total
  float* ws     = (float*)d_ws;
  float* ws_xs  = ws;                                        // 5000*128
  float* ws_xv  = ws_xs + (size_t)N_ATOMS * FDIM;            // 5000*384
  float* ws_h   = ws_xv + (size_t)N_ATOMS * 3 * FDIM;        // 50000*128
  float* ws_W   = ws_h  + (size_t)N_EDGES * FDIM;            // 50000*512
  float* ws_agg = ws_W  + (size_t)N_EDGES * 512;             // 5000*1024
  float* ws_S1  = ws_agg + (size_t)N_ATOMS * 1024;           // 5000*128
  float* ws_V1  = ws_S1 + (size_t)N_ATOMS * CDIM;            // 5000*384
  int*   ws_st  = (int*)(ws_V1 + (size_t)N_ATOMS * 3 * CDIM);// 5001

  const dim3 blk(256);
  const int gA = (N_ATOMS + 127) / 128;   // 40
  const int gE = (N_EDGES + 127) / 128;   // 391

  // 1) xs = x[:, :128] @ W_in2f_s / sqrt(128)
  gemm_wmma_f32<1, CDIM><<<dim3(gA, 2), blk, 0, stream>>>(
      x, 4 * CDIM, W_in2f_s, FDIM, nullptr, nullptr,
      ws_xs, FDIM, 1, N_ATOMS, INV_SQRT128_C, 0);

  // 2) xv[n,f,i], strided A (x[:,128:] viewed as [n,128,3]) -> stored [n,f,i]
  for (int i = 0; i < 3; ++i) {
    gemm_wmma_f32<3, CDIM><<<dim3(gA, 2), blk, 0, stream>>>(
        x + CDIM + i, 4 * CDIM, W_in2f_v, FDIM, nullptr, nullptr,
        ws_xv + i, 3 * FDIM, 3, N_ATOMS, INV_SQRT128_C, 0);
  }

  // 3) h = ssp(f_ij @ Wf1 + bf1), K=20
  gemm_wmma_f32<1, NRBF><<<dim3(gE, 2), blk, 0, stream>>>(
      f_ij, NRBF, Wf1, FDIM, bf1, nullptr,
      ws_h, FDIM, 1, N_EDGES, 1.0f, 1);

  // 4) W = (h @ Wf2[:, :512] + bf2[:512]) * rcut
  gemm_wmma_f32<1, FDIM><<<dim3(gE, 8), blk, 0, stream>>>(
      ws_h, FDIM, Wf2, 6 * FDIM, bf2, rcut_ij,
      ws_W, 512, 1, N_EDGES, 1.0f, 0);

  // 5) CSR starts from sorted idx_i
  build_starts<<<dim3((N_ATOMS + 1 + 127) / 128), dim3(128), 0, stream>>>(
      idx_i, N_EDGES, N_ATOMS, ws_st);

  // 6) fused gather + featurize + filter + deterministic segment sum
  edge_aggregate<<<dim3(N_ATOMS), dim3(128), 0, stream>>>(
      ws_st, idx_j, r_ij, ws_xs, ws_xv, ws_W, ws_agg);

  // 7) S1 = ssp(agg[:, :256] @ W1_s / 16)
  gemm_wmma_f32<1, 2 * FDIM><<<dim3(gA, 2), blk, 0, stream>>>(
      ws_agg, 1024, W1_s, CDIM, nullptr, nullptr,
      ws_S1, CDIM, 1, N_ATOMS, 0.0625f, 1);

  // 8) V1[n,c,i] = agg[:, 256:1024] as [n,256,3] contracted with W1_v / 16
  for (int i = 0; i < 3; ++i) {
    gemm_wmma_f32<3, 2 * FDIM><<<dim3(gA, 2), blk, 0, stream>>>(
        ws_agg + 256 + i, 1024, W1_v, CDIM, nullptr, nullptr,
        ws_V1 + i, 3 * CDIM, 3, N_ATOMS, 0.0625f, 0);
  }

  // 9) out[:, :128] = S1 @ W2_s / sqrt(128)
  gemm_wmma_f32<1, CDIM><<<dim3(gA, 2), blk, 0, stream>>>(
      ws_S1, CDIM, W2_s, CDIM, nullptr, nullptr,
      out, 4 * CDIM, 1, N_ATOMS, INV_SQRT128_C, 0);

  // 10) out[:, 128+3c+i] = V1[:, :, i] @ W2_v / sqrt(128)
  for (int i = 0; i < 3; ++i) {
    gemm_wmma_f32<3, CDIM><<<dim3(gA, 2), blk, 0, stream>>>(
        ws_V1 + i, 3 * CDIM, W2_v, CDIM, nullptr, nullptr,
        out + CDIM + i, 4 * CDIM, 3, N_ATOMS, INV_SQRT128_C, 0);
  }
}